// PDQP_Net_AR_3582002725324
// MI455X (gfx1250) — compile-verified
//
#include <hip/hip_runtime.h>
#include <hip/hip_bf16.h>
#include <math.h>

typedef float v2f __attribute__((ext_vector_type(2)));
typedef float v8f __attribute__((ext_vector_type(8)));

#define NN 4096
#define FEAT 64
#define NF (NN * FEAT)
#define SA 68   // LDS stride (floats) for A tile: bank = 4*lane + k -> conflict-free
#define SB 80   // LDS stride (floats) for B tile: half-waves hit disjoint bank halves

// ---------------------------------------------------------------------------
// init: X0 = x*Wix^T + bix  -> XC, XP, XB ; Y0 = y*Wiy^T + biy -> YC
// ---------------------------------------------------------------------------
__global__ __launch_bounds__(256) void init_kernel(
    const float* __restrict__ x, const float* __restrict__ y,
    const float* __restrict__ Wix, const float* __restrict__ bix,
    const float* __restrict__ Wiy, const float* __restrict__ biy,
    float* __restrict__ XC, float* __restrict__ XP, float* __restrict__ XB,
    float* __restrict__ YC)
{
    int idx = blockIdx.x * 256 + threadIdx.x;  // 0 .. 2*NF-1
    int f = idx & 63;
    if (idx < NF) {
        int i = idx >> 6;
        float v = x[i] * Wix[f] + bix[f];
        XC[idx] = v; XP[idx] = v; XB[idx] = v;
    } else {
        int j = idx - NF;
        int i = j >> 6;
        YC[j] = y[i] * Wiy[f] + biy[f];
    }
}

// ---------------------------------------------------------------------------
// prep: xd = theta*(xc-xp)+xc ; xmd = (1-beta)*xb + beta*xc
// ---------------------------------------------------------------------------
__global__ __launch_bounds__(256) void prep_kernel(
    const float* __restrict__ XP, const float* __restrict__ XC,
    const float* __restrict__ XB,
    const float* __restrict__ beta_a, const float* __restrict__ theta_a, int layer,
    float* __restrict__ Xd, float* __restrict__ Xmd)
{
    int idx = blockIdx.x * 256 + threadIdx.x;
    float beta = beta_a[layer], theta = theta_a[layer];
    float xc = XC[idx], xp = XP[idx], xb = XB[idx];
    Xd[idx]  = theta * (xc - xp) + xc;
    Xmd[idx] = (1.0f - beta) * xb + beta * xc;
}

// ---------------------------------------------------------------------------
// gemm3: one launch computes T1 = A@Xd, T2 = Q@Xmd, T3 = AT@Yc
// 192 blocks (64 per product), 128 threads = 4 waves, 64 output rows / block.
// f32 WMMA 16x16x4; software-pipelined: next global tile is prefetched into
// registers while the current LDS tile feeds the matrix units.
// ---------------------------------------------------------------------------
__global__ __launch_bounds__(128) void gemm3_kernel(
    const float* __restrict__ A, const float* __restrict__ Q,
    const float* __restrict__ AT,
    const float* __restrict__ Xd, const float* __restrict__ Xmd,
    const float* __restrict__ Yc,
    float* __restrict__ T1, float* __restrict__ T2, float* __restrict__ T3)
{
    __shared__ __align__(16) float As[64 * SA];
    __shared__ __align__(16) float Bs[64 * SB];

    int bid = blockIdx.x;
    int which = bid >> 6;
    int rb = (bid & 63) * 64;
    const float* M_; const float* V_; float* O_;
    if (which == 0)      { M_ = A;  V_ = Xd;  O_ = T1; }
    else if (which == 1) { M_ = Q;  V_ = Xmd; O_ = T2; }
    else                 { M_ = AT; V_ = Yc;  O_ = T3; }

    int tid   = threadIdx.x;
    int wave  = tid >> 5;
    int lane  = tid & 31;
    int lhalf = lane >> 4;   // 0/1 selects K pair
    int lq    = lane & 15;   // M (for A frag) / N (for B frag, C/D cols)

    // per-thread staging offsets (8 float4 each for A and B tiles)
    int sr[8], sc[8];
    #pragma unroll
    for (int t = 0; t < 8; ++t) {
        int idx = t * 128 + tid;          // 0..1023
        sr[t] = idx >> 4;                 // tile row 0..63
        sc[t] = (idx & 15) << 2;          // tile col (float4 granularity)
    }

    v8f acc0 = {0,0,0,0,0,0,0,0};
    v8f acc1 = {0,0,0,0,0,0,0,0};
    v8f acc2 = {0,0,0,0,0,0,0,0};
    v8f acc3 = {0,0,0,0,0,0,0,0};

    float4 rA[8], rB[8];

    // prologue: fetch tile kb=0 and commit to LDS
    #pragma unroll
    for (int t = 0; t < 8; ++t) {
        rA[t] = *(const float4*)&M_[(size_t)(rb + sr[t]) * NN + sc[t]];
        rB[t] = *(const float4*)&V_[(size_t)sr[t] * FEAT + sc[t]];
    }
    #pragma unroll
    for (int t = 0; t < 8; ++t) {
        float* da = &As[sr[t] * SA + sc[t]];
        da[0] = rA[t].x; da[1] = rA[t].y; da[2] = rA[t].z; da[3] = rA[t].w;
        float* db = &Bs[sr[t] * SB + sc[t]];
        db[0] = rB[t].x; db[1] = rB[t].y; db[2] = rB[t].z; db[3] = rB[t].w;
    }
    __syncthreads();

    const float* arow = &As[(wave * 16 + lq) * SA + (lhalf << 1)];

    for (int kb = 0; kb < NN; kb += 64) {
        int kn = kb + 64;
        bool hasNext = (kn < NN);
        // prefetch next tile into registers (overlaps with WMMA below)
        if (hasNext) {
            #pragma unroll
            for (int t = 0; t < 8; ++t) {
                rA[t] = *(const float4*)&M_[(size_t)(rb + sr[t]) * NN + kn + sc[t]];
                rB[t] = *(const float4*)&V_[(size_t)(kn + sr[t]) * FEAT + sc[t]];
            }
        }
        // compute on current LDS tile
        #pragma unroll
        for (int kk = 0; kk < 64; kk += 4) {
            v2f a = *(const v2f*)(arow + kk);                 // A[m][kk+koff .. +1]
            const float* b0 = &Bs[(kk + (lhalf << 1)) * SB + lq];
            v2f b;
            b.x = b0[0];       b.y = b0[SB];
            acc0 = __builtin_amdgcn_wmma_f32_16x16x4_f32(false, a, false, b, (short)0, acc0, false, false);
            b.x = b0[16];      b.y = b0[SB + 16];
            acc1 = __builtin_amdgcn_wmma_f32_16x16x4_f32(false, a, false, b, (short)0, acc1, false, false);
            b.x = b0[32];      b.y = b0[SB + 32];
            acc2 = __builtin_amdgcn_wmma_f32_16x16x4_f32(false, a, false, b, (short)0, acc2, false, false);
            b.x = b0[48];      b.y = b0[SB + 48];
            acc3 = __builtin_amdgcn_wmma_f32_16x16x4_f32(false, a, false, b, (short)0, acc3, false, false);
        }
        __syncthreads();   // all waves done reading the tile
        if (hasNext) {
            #pragma unroll
            for (int t = 0; t < 8; ++t) {
                float* da = &As[sr[t] * SA + sc[t]];
                da[0] = rA[t].x; da[1] = rA[t].y; da[2] = rA[t].z; da[3] = rA[t].w;
                float* db = &Bs[sr[t] * SB + sc[t]];
                db[0] = rB[t].x; db[1] = rB[t].y; db[2] = rB[t].z; db[3] = rB[t].w;
            }
        }
        __syncthreads();
    }

    // C/D layout: VGPR j -> (M=j, lanes 0-15) / (M=j+8, lanes 16-31), N = lane&15
    int rbase = rb + wave * 16 + lhalf * 8;
    #pragma unroll
    for (int j = 0; j < 8; ++j) {
        float* orow = &O_[(size_t)(rbase + j) * FEAT + lq];
        orow[0]  = acc0[j];
        orow[16] = acc1[j];
        orow[32] = acc2[j];
        orow[48] = acc3[j];
    }
}

// ---------------------------------------------------------------------------
// yupd: YN = proj( YC + gamma*(T1 @ W1^T) - b , Iy )
// ---------------------------------------------------------------------------
__global__ __launch_bounds__(256) void yupd_kernel(
    const float* __restrict__ T1, const float* __restrict__ YC,
    const float* __restrict__ W1, const float* __restrict__ b,
    const float* __restrict__ Iy,
    const float* __restrict__ gamma_a, int layer,
    float* __restrict__ YN)
{
    __shared__ float W1s[64 * 65];
    __shared__ float rT[4 * 64];
    int tid = threadIdx.x, f = tid & 63, r = tid >> 6;
    const float* W1l = W1 + (size_t)layer * 64 * 64;
    for (int t = tid; t < 4096; t += 256) W1s[(t >> 6) * 65 + (t & 63)] = W1l[t];
    float gamma = gamma_a[layer];
    int rbq = blockIdx.x * 64;
    __syncthreads();
    for (int it = 0; it < 16; ++it) {
        int row = rbq + it * 4 + r;
        rT[r * 64 + f] = T1[(size_t)row * 64 + f];
        __syncthreads();
        float s = 0.f;
        #pragma unroll
        for (int k = 0; k < 64; ++k) s += rT[r * 64 + k] * W1s[f * 65 + k];
        float val = YC[(size_t)row * 64 + f] + gamma * s - b[row];
        YN[(size_t)row * 64 + f] = val + Iy[row] * fmaxf(-val, 0.f);
        __syncthreads();
    }
}

// ---------------------------------------------------------------------------
// xupdA: Tn = XC@W2^T - eta*(T2@W3^T + b3 + c + T3@W4^T + b4)
// ---------------------------------------------------------------------------
__global__ __launch_bounds__(256) void xupdA_kernel(
    const float* __restrict__ XC, const float* __restrict__ T2,
    const float* __restrict__ T3,
    const float* __restrict__ W2, const float* __restrict__ W3,
    const float* __restrict__ W4,
    const float* __restrict__ b3, const float* __restrict__ b4,
    const float* __restrict__ c,
    const float* __restrict__ eta_a, int layer,
    float* __restrict__ Tn)
{
    __shared__ float W2s[64 * 65], W3s[64 * 65], W4s[64 * 65];
    __shared__ float rX[4 * 64], r2[4 * 64], r3[4 * 64];
    int tid = threadIdx.x, f = tid & 63, r = tid >> 6;
    size_t off = (size_t)layer * 64 * 64;
    for (int t = tid; t < 4096; t += 256) {
        int ff = t >> 6, kk = t & 63;
        W2s[ff * 65 + kk] = W2[off + t];
        W3s[ff * 65 + kk] = W3[off + t];
        W4s[ff * 65 + kk] = W4[off + t];
    }
    float eta = eta_a[layer];
    float b3f = b3[layer * 64 + f], b4f = b4[layer * 64 + f];
    int rbq = blockIdx.x * 64;
    __syncthreads();
    for (int it = 0; it < 16; ++it) {
        int row = rbq + it * 4 + r;
        size_t ro = (size_t)row * 64 + f;
        rX[r * 64 + f] = XC[ro]; r2[r * 64 + f] = T2[ro]; r3[r * 64 + f] = T3[ro];
        __syncthreads();
        float s2 = 0.f, s3 = 0.f, s4 = 0.f;
        #pragma unroll
        for (int k = 0; k < 64; ++k) {
            s2 += rX[r * 64 + k] * W2s[f * 65 + k];
            s3 += r2[r * 64 + k] * W3s[f * 65 + k];
            s4 += r3[r * 64 + k] * W4s[f * 65 + k];
        }
        Tn[ro] = s2 - eta * (s3 + b3f + c[row] + s4 + b4f);
        __syncthreads();
    }
}

// ---------------------------------------------------------------------------
// xupdB: xuu = t - iu*relu([t,u]@Wp1^T) ; xf = xuu + il*relu([xuu,l]@Wp2^T)
//        XN = xf ; XBN = (1-beta)*XB + beta*xf
// ---------------------------------------------------------------------------
__global__ __launch_bounds__(256) void xupdB_kernel(
    const float* __restrict__ Tn, const float* __restrict__ XBold,
    const float* __restrict__ Wp1, const float* __restrict__ Wp2,
    const float* __restrict__ iu, const float* __restrict__ il,
    const float* __restrict__ u, const float* __restrict__ l,
    const float* __restrict__ beta_a, int layer,
    float* __restrict__ XN, float* __restrict__ XBN)
{
    __shared__ float P1s[64 * 66], P2s[64 * 66];
    __shared__ float rT[4 * 64], rU[4 * 64];
    int tid = threadIdx.x, f = tid & 63, r = tid >> 6;
    size_t off = (size_t)layer * 64 * 65;
    for (int t = tid; t < 64 * 65; t += 256) {
        int ff = t / 65, kk = t - ff * 65;
        P1s[ff * 66 + kk] = Wp1[off + t];
        P2s[ff * 66 + kk] = Wp2[off + t];
    }
    float beta = beta_a[layer];
    int rbq = blockIdx.x * 64;
    __syncthreads();
    for (int it = 0; it < 16; ++it) {
        int row = rbq + it * 4 + r;
        size_t ro = (size_t)row * 64 + f;
        rT[r * 64 + f] = Tn[ro];
        __syncthreads();
        float s1 = u[row] * P1s[f * 66 + 64];
        #pragma unroll
        for (int k = 0; k < 64; ++k) s1 += rT[r * 64 + k] * P1s[f * 66 + k];
        float xuu = rT[r * 64 + f] - iu[row] * fmaxf(s1, 0.f);
        rU[r * 64 + f] = xuu;
        __syncthreads();
        float s2 = l[row] * P2s[f * 66 + 64];
        #pragma unroll
        for (int k = 0; k < 64; ++k) s2 += rU[r * 64 + k] * P2s[f * 66 + k];
        float xf = xuu + il[row] * fmaxf(s2, 0.f);
        XN[ro]  = xf;
        XBN[ro] = (1.0f - beta) * XBold[ro] + beta * xf;
        __syncthreads();
    }
}

// ---------------------------------------------------------------------------
// outxy: x_out = XC@Wox^T + box -> out[0..N) ; y_out -> out[N..2N)
// ---------------------------------------------------------------------------
__global__ __launch_bounds__(256) void outxy_kernel(
    const float* __restrict__ XC, const float* __restrict__ YC,
    const float* __restrict__ Wox, const float* __restrict__ box,
    const float* __restrict__ Woy, const float* __restrict__ boy,
    float* __restrict__ out)
{
    int idx = blockIdx.x * 256 + threadIdx.x;  // 0 .. 2N-1
    if (idx < NN) {
        float s = box[0];
        #pragma unroll 8
        for (int f = 0; f < 64; ++f) s += XC[(size_t)idx * 64 + f] * Wox[f];
        out[idx] = s;
    } else {
        int j = idx - NN;
        float s = boy[0];
        #pragma unroll 8
        for (int f = 0; f < 64; ++f) s += YC[(size_t)j * 64 + f] * Woy[f];
        out[NN + j] = s;
    }
}

// ---------------------------------------------------------------------------
// matvec3: Ax = A@xo, Qx = Q@xo, ATy = AT@yo.  One wave per row.
// ---------------------------------------------------------------------------
__global__ __launch_bounds__(256) void matvec3_kernel(
    const float* __restrict__ A, const float* __restrict__ Q,
    const float* __restrict__ AT,
    const float* __restrict__ xo, const float* __restrict__ yo,
    float* __restrict__ Ax, float* __restrict__ Qx, float* __restrict__ ATy)
{
    int gwave = (blockIdx.x * 256 + threadIdx.x) >> 5;  // 0 .. 3*N-1
    int lane = threadIdx.x & 31;
    int which = gwave >> 12;
    int row = gwave & (NN - 1);
    const float* M_ = (which == 0) ? A : (which == 1) ? Q : AT;
    const float* v_ = (which == 2) ? yo : xo;
    const float4* a4 = (const float4*)&M_[(size_t)row * NN];
    const float4* v4 = (const float4*)v_;
    float s = 0.f;
    for (int k = lane; k < NN / 4; k += 32) {
        float4 a = a4[k], v = v4[k];
        s += a.x * v.x + a.y * v.y + a.z * v.z + a.w * v.w;
    }
    #pragma unroll
    for (int off = 16; off > 0; off >>= 1) s += __shfl_down(s, off, 32);
    if (lane == 0) {
        float* o = (which == 0) ? Ax : (which == 1) ? Qx : ATy;
        o[row] = s;
    }
}

// ---------------------------------------------------------------------------
// reduce1: 7 partial sums per block (deterministic LDS tree). 16 blocks.
// q0=|proj(Ax-b)|^2 q1=|b|^2 q2=|Qx+ATy+c|^2 q3=|c|^2 q4=x.Qx q5=c.x q6=b.y
// ---------------------------------------------------------------------------
__global__ __launch_bounds__(256) void reduce1_kernel(
    const float* __restrict__ Ax, const float* __restrict__ Qx,
    const float* __restrict__ ATy,
    const float* __restrict__ xo, const float* __restrict__ yo,
    const float* __restrict__ b, const float* __restrict__ c,
    const float* __restrict__ Iy,
    float* __restrict__ partials)
{
    __shared__ float red[7 * 256];
    int tid = threadIdx.x;
    int i = blockIdx.x * 256 + tid;  // 0..4095
    float av = Ax[i] - b[i];
    float pr = av + Iy[i] * fmaxf(-av, 0.f);
    float kk = Qx[i] + ATy[i] + c[i];
    float s[7];
    s[0] = pr * pr; s[1] = b[i] * b[i]; s[2] = kk * kk; s[3] = c[i] * c[i];
    s[4] = xo[i] * Qx[i]; s[5] = c[i] * xo[i]; s[6] = b[i] * yo[i];
    #pragma unroll
    for (int q = 0; q < 7; ++q) red[q * 256 + tid] = s[q];
    __syncthreads();
    for (int off = 128; off > 0; off >>= 1) {
        if (tid < off) {
            #pragma unroll
            for (int q = 0; q < 7; ++q) red[q * 256 + tid] += red[q * 256 + tid + off];
        }
        __syncthreads();
    }
    if (tid < 7) partials[blockIdx.x * 7 + tid] = red[tid * 256];
}

__global__ void finalize_kernel(const float* __restrict__ partials,
                                float* __restrict__ out)
{
    if (threadIdx.x == 0 && blockIdx.x == 0) {
        float s[7] = {0, 0, 0, 0, 0, 0, 0};
        for (int bkt = 0; bkt < 16; ++bkt)
            for (int q = 0; q < 7; ++q) s[q] += partials[bkt * 7 + q];
        float eps = 0.0001f;
        float t1 = sqrtf(s[0]) / (eps + sqrtf(s[1]));
        float t2 = sqrtf(s[2]) / (eps + sqrtf(s[3]));
        float t3 = fabsf(s[4] + s[5] + s[6]);
        out[2 * NN] = t1 + t2 + t3;
    }
}

// ---------------------------------------------------------------------------
extern "C" void kernel_launch(void* const* d_in, const int* in_sizes, int n_in,
                              void* d_out, int out_size, void* d_ws, size_t ws_size,
                              hipStream_t stream)
{
    (void)in_sizes; (void)n_in; (void)out_size; (void)ws_size;
    const float* A    = (const float*)d_in[0];
    const float* Q    = (const float*)d_in[1];
    const float* b    = (const float*)d_in[2];
    const float* c    = (const float*)d_in[3];
    const float* x    = (const float*)d_in[4];
    const float* y    = (const float*)d_in[5];
    const float* Iy   = (const float*)d_in[6];
    const float* il   = (const float*)d_in[7];   // indicator_x_l
    const float* iu   = (const float*)d_in[8];   // indicator_x_u
    const float* l    = (const float*)d_in[9];
    const float* u    = (const float*)d_in[10];
    const float* Wix  = (const float*)d_in[11];
    const float* bix  = (const float*)d_in[12];
    const float* Wiy  = (const float*)d_in[13];
    const float* biy  = (const float*)d_in[14];
    const float* Wox  = (const float*)d_in[15];
    const float* box  = (const float*)d_in[16];
    const float* Woy  = (const float*)d_in[17];
    const float* boy  = (const float*)d_in[18];
    const float* W1   = (const float*)d_in[19];
    const float* W2   = (const float*)d_in[20];
    const float* W3   = (const float*)d_in[21];
    const float* b3   = (const float*)d_in[22];
    const float* W4   = (const float*)d_in[23];
    const float* b4   = (const float*)d_in[24];
    const float* Wp1  = (const float*)d_in[25];
    const float* Wp2  = (const float*)d_in[26];
    const float* gamma_a = (const float*)d_in[27];
    const float* eta_a   = (const float*)d_in[28];
    const float* beta_a  = (const float*)d_in[29];
    const float* theta_a = (const float*)d_in[30];
    const float* AT   = (const float*)d_in[31];

    float* ws = (float*)d_ws;
    float* pool[7];
    for (int i = 0; i < 7; ++i) pool[i] = ws + (size_t)i * NF;
    float* Xd  = ws + (size_t)7  * NF;
    float* Xmd = ws + (size_t)8  * NF;
    float* T1  = ws + (size_t)9  * NF;
    float* T2  = ws + (size_t)10 * NF;
    float* T3  = ws + (size_t)11 * NF;
    float* Ax  = ws + (size_t)12 * NF;
    float* Qx  = Ax + NN;
    float* ATy = Qx + NN;
    float* partials = ATy + NN;

    float *XP = pool[0], *XC = pool[1], *XB = pool[2], *YC = pool[3];
    float *F0 = pool[4], *F1 = pool[5], *F2 = pool[6];

    init_kernel<<<(2 * NF) / 256, 256, 0, stream>>>(x, y, Wix, bix, Wiy, biy,
                                                    XC, XP, XB, YC);
    for (int layer = 0; layer < 8; ++layer) {
        prep_kernel<<<NF / 256, 256, 0, stream>>>(XP, XC, XB, beta_a, theta_a,
                                                  layer, Xd, Xmd);
        gemm3_kernel<<<192, 128, 0, stream>>>(A, Q, AT, Xd, Xmd, YC, T1, T2, T3);
        yupd_kernel<<<64, 256, 0, stream>>>(T1, YC, W1, b, Iy, gamma_a, layer, F0);
        xupdA_kernel<<<64, 256, 0, stream>>>(XC, T2, T3, W2, W3, W4, b3, b4, c,
                                             eta_a, layer, Xmd /* Tn */);
        xupdB_kernel<<<64, 256, 0, stream>>>(Xmd, XB, Wp1, Wp2, iu, il, u, l,
                                             beta_a, layer, F1, F2);
        // rotate carries: (xp,xc,xb,yc) <- (xc, xn, xbn, yn)
        float *nXP = XC, *nXC = F1, *nXB = F2, *nYC = F0;
        F0 = XP; F1 = XB; F2 = YC;
        XP = nXP; XC = nXC; XB = nXB; YC = nYC;
    }

    float* out = (float*)d_out;
    outxy_kernel<<<(2 * NN) / 256, 256, 0, stream>>>(XC, YC, Wox, box, Woy, boy, out);
    matvec3_kernel<<<(3 * NN) / 8, 256, 0, stream>>>(A, Q, AT, out, out + NN,
                                                     Ax, Qx, ATy);
    reduce1_kernel<<<16, 256, 0, stream>>>(Ax, Qx, ATy, out, out + NN, b, c, Iy,
                                           partials);
    finalize_kernel<<<1, 32, 0, stream>>>(partials, out);
}